// PtsManipulator_76613626626605
// MI455X (gfx1250) — compile-verified
//
#include <hip/hip_runtime.h>
#include <stdint.h>

#define BSZ 8
#define CH  3
#define SS  512
#define NN  (SS*SS)
#define EPSV 0.01f
#define PPPK 8
#define SENT 0xFFFFFFFFFFFFFFFFull

typedef __attribute__((ext_vector_type(2))) float v2f;
typedef __attribute__((ext_vector_type(8))) float v8f;

// Mcomb[b] = K[b] * RTinv_cam2[b] * RT_cam1[b] * K_inv[b]   (4x4 each)
__global__ void mcomb_kernel(const float* __restrict__ K, const float* __restrict__ Kinv,
                             const float* __restrict__ RT1, const float* __restrict__ RTinv2,
                             float* __restrict__ M) {
    int t = threadIdx.x;
    if (t >= BSZ * 16) return;
    int b = t >> 4, e = t & 15, i = e >> 2, j = e & 3;
    const float* Kb  = K      + b * 16;
    const float* Kib = Kinv   + b * 16;
    const float* R1  = RT1    + b * 16;
    const float* R2i = RTinv2 + b * 16;
    float acc = 0.f;
    for (int a = 0; a < 4; ++a)
        for (int p = 0; p < 4; ++p)
            for (int q = 0; q < 4; ++q)
                acc += Kb[i*4+a] * R2i[a*4+p] * R1[p*4+q] * Kib[q*4+j];
    M[b * 16 + e] = acc;
}

// One wave = 32 points via two V_WMMA_F32_16X16X4_F32:
//   D = A(16x4 = Mcomb rows 0..3, padded) x B(4x16 = 16 point columns)
// Lane L<16 gets point L's (x',y',z') in D[0..2]. The second WMMA's lanes
// 0-15 are routed to lanes 16-31 with ds_bpermute executed in UNIFORM control
// flow (full EXEC; disabled-lane reads would return 0), then cndmask-selected.
__global__ __launch_bounds__(256) void project_splat_kernel(
    const float* __restrict__ pred, const float* __restrict__ M,
    unsigned long long* __restrict__ slots) {
    const int  lane = threadIdx.x & 31;
    const int  wave = (blockIdx.x * blockDim.x + threadIdx.x) >> 5;
    const int  sub  = lane & 15;
    const bool hi   = lane >= 16;
    const int  p0   = wave * 32;

    const float sc = 2.0f / (SS - 1.0f);
    const int ptA = p0 + sub,      ptB = p0 + 16 + sub;
    const float dA = pred[ptA],    dB = pred[ptB];
    const float xxA = (ptA & (SS-1)) * sc - 1.0f, yyA = (ptA >> 9) * sc - 1.0f;
    const float xxB = (ptB & (SS-1)) * sc - 1.0f, yyB = (ptB >> 9) * sc - 1.0f;

    // B (4x16): lanes 0-15 hold rows K=0,1; lanes 16-31 rows K=2,3.
    v2f bmA; bmA[0] = hi ? -dA : xxA * dA;  bmA[1] = hi ? 1.0f : -yyA * dA;
    v2f bmB; bmB[0] = hi ? -dB : xxB * dB;  bmB[1] = hi ? 1.0f : -yyB * dB;
    // A (16x4): lane m holds A[m][0],A[m][1] (lo) / A[m][2],A[m][3] (hi); rows>=4 zero.
    v2f am;
    am[0] = (sub < 4) ? M[sub * 4 + (hi ? 2 : 0)] : 0.0f;
    am[1] = (sub < 4) ? M[sub * 4 + (hi ? 3 : 1)] : 0.0f;

    const v8f cz = {0.f,0.f,0.f,0.f,0.f,0.f,0.f,0.f};
    v8f acc1 = __builtin_amdgcn_wmma_f32_16x16x4_f32(false, am, false, bmA,
                                                     (short)0, cz, false, false);
    v8f acc2 = __builtin_amdgcn_wmma_f32_16x16x4_f32(false, am, false, bmB,
                                                     (short)0, cz, false, false);

    // Unconditional, full-EXEC lane routing: every lane gathers lane (lane&15).
    const int bidx = sub * 4;
    const float pX = __int_as_float(__builtin_amdgcn_ds_bpermute(bidx, __float_as_int(acc2[0])));
    const float pY = __int_as_float(__builtin_amdgcn_ds_bpermute(bidx, __float_as_int(acc2[1])));
    const float pZ = __int_as_float(__builtin_amdgcn_ds_bpermute(bidx, __float_as_int(acc2[2])));
    const float X = hi ? pX : acc1[0];
    const float Y = hi ? pY : acc1[1];
    const float Z = hi ? pZ : acc1[2];
    const int   pt = p0 + lane;

    const bool  msk = fabsf(Z) < EPSV;
    const float zs  = msk ? EPSV : Z;
    const float t   = 1.0f / (-zs);            // single divide, never 0
    const float x   = msk ? -10.f : X * t;                 // * +1
    const float y   = -(msk ? -10.f : Y * t);              // * -1
    const float z   = -(msk ? -10.f : zs);                 // * -1
    const float yi  = -y;
    const float ixf = rintf((x  + 1.f) * 0.5f * (SS - 1.f));
    const float iyf = rintf((yi + 1.f) * 0.5f * (SS - 1.f));
    const bool valid = (fabsf(x) <= 1.f) && (fabsf(yi) <= 1.f) && (z > 0.f);
    if (valid) {
        const int ixi = (int)fminf(fmaxf(ixf, 0.f), SS - 1.f);
        const int iyi = (int)fminf(fmaxf(iyf, 0.f), SS - 1.f);
        const int pix = iyi * SS + ixi;
        unsigned long long cur =
            ((unsigned long long)__float_as_uint(z) << 32) | (unsigned)pt;
        unsigned long long* sp = slots + (size_t)pix * PPPK;
        #pragma unroll
        for (int s = 0; s < PPPK; ++s) {
            unsigned long long old = atomicMin(&sp[s], cur);
            if (old == SENT) break;          // landed in empty slot
            cur = old > cur ? old : cur;     // carry the loser downward
        }
    }
}

// One thread per pixel: composite the (sorted) up-to-8 nearest points,
// then restore the sentinel so the next batch / next replay starts clean.
__global__ __launch_bounds__(256) void composite_kernel(
    unsigned long long* __restrict__ slots, const float* __restrict__ pred,
    const float* __restrict__ M, const float* __restrict__ src,
    float* __restrict__ out) {
    const int pix = blockIdx.x * blockDim.x + threadIdx.x;
    if (pix >= NN) return;
    const float sc = 2.0f / (SS - 1.0f);
    const int ixi = pix & (SS - 1), iyi = pix >> 9;
    const float px =  ixi * sc - 1.f;
    const float py = -(iyi * sc - 1.f);
    const float m00 = M[0], m01 = M[1], m02 = M[2], m03 = M[3];
    const float m10 = M[4], m11 = M[5], m12 = M[6], m13 = M[7];
    const float inv_rad2 = (512.0f / 3.0f) * (512.0f / 3.0f);  // 1 / RAD_NDC^2

    // Load all 8 slots with 4x b128.
    ulonglong2* sp = (ulonglong2*)(slots + (size_t)pix * PPPK);
    ulonglong2 q0 = sp[0], q1 = sp[1], q2 = sp[2], q3 = sp[3];
    const unsigned long long v[PPPK] = { q0.x, q0.y, q1.x, q1.y,
                                         q2.x, q2.y, q3.x, q3.y };

    float T = 1.f, o0 = 0.f, o1 = 0.f, o2 = 0.f;
    #pragma unroll
    for (int s = 0; s < PPPK; ++s) {
        if (v[s] == SENT) break;
        const int   idx = (int)(unsigned)(v[s] & 0xFFFFFFFFu);
        const float z   = __uint_as_float((unsigned)(v[s] >> 32));   // z > 0
        const float dd  = pred[idx];
        const float p0  =  ((idx & (SS-1)) * sc - 1.f) * dd;
        const float p1  = -((idx >> 9)     * sc - 1.f) * dd;
        const float p2  = -dd;
        const float X = m00 * p0 + m01 * p1 + m02 * p2 + m03;
        const float Y = m10 * p0 + m11 * p1 + m12 * p2 + m13;
        const float tz = 1.0f / z;
        const float x  =  X * tz;      // X / (-zs), zs = -z
        const float y  = -(Y * tz);
        const float dx = x - px, dy = y - py;
        float dist = (dx * dx + dy * dy) * inv_rad2;
        dist = fminf(fmaxf(dist, 0.001f), 1.0f);
        const float alpha = 1.0f - dist;       // TAU == 1
        const float w = alpha * T;
        o0 += w * src[0 * NN + idx];
        o1 += w * src[1 * NN + idx];
        o2 += w * src[2 * NN + idx];
        T *= (1.0f - fminf(alpha, 0.9999f));   // log1p(-clip) product
    }
    out[0 * NN + pix] = o0;
    out[1 * NN + pix] = o1;
    out[2 * NN + pix] = o2;

    const ulonglong2 sent2 = { SENT, SENT };
    sp[0] = sent2; sp[1] = sent2; sp[2] = sent2; sp[3] = sent2;
}

extern "C" void kernel_launch(void* const* d_in, const int* in_sizes, int n_in,
                              void* d_out, int out_size, void* d_ws, size_t ws_size,
                              hipStream_t stream) {
    const float* src    = (const float*)d_in[0];  // (8,3,512,512)
    const float* pred   = (const float*)d_in[1];  // (8,1,512,512)
    const float* K      = (const float*)d_in[2];
    const float* Kinv   = (const float*)d_in[3];
    const float* RT1    = (const float*)d_in[4];
    const float* RTinv2 = (const float*)d_in[7];
    float* out = (float*)d_out;

    float* mcomb = (float*)d_ws;                               // 512 B used
    unsigned long long* slots =
        (unsigned long long*)((char*)d_ws + 1024);             // 16 MB: NN * 8 * u64

    mcomb_kernel<<<1, 128, 0, stream>>>(K, Kinv, RT1, RTinv2, mcomb);

    // One-time sentinel init per call; composite restores sentinels per batch.
    hipMemsetAsync(slots, 0xFF, (size_t)NN * PPPK * sizeof(unsigned long long), stream);

    for (int b = 0; b < BSZ; ++b) {
        project_splat_kernel<<<NN / 256, 256, 0, stream>>>(
            pred + (size_t)b * NN, mcomb + b * 16, slots);
        composite_kernel<<<NN / 256, 256, 0, stream>>>(
            slots, pred + (size_t)b * NN, mcomb + b * 16,
            src + (size_t)b * CH * NN, out + (size_t)b * CH * NN);
    }
}